// MultitaskSNN_31104153157959
// MI455X (gfx1250) — compile-verified
//
#include <hip/hip_runtime.h>
#include <hip/hip_bf16.h>

// ---------------------------------------------------------------------------
// Multitask LIF SNN on MI455X (gfx1250), wave32.
//
// Strategy: each wave owns a 16-batch tile; loops t=0..99 with all LIF
// membrane state resident in WMMA C/D-layout VGPRs. All matvecs are done
// with V_WMMA_F32_16X16X4_F32 (weights as A, batch as N, K chunked by 4,
// bias folded in as an augmented K row). Spike matrices are re-striped from
// D-layout to B-layout with ds_bpermute_b32 pairs.
// ---------------------------------------------------------------------------

typedef __attribute__((ext_vector_type(2))) float v2f;
typedef __attribute__((ext_vector_type(8))) float v8f;

#define TSTEPS 100
#define BATCH  16384

__device__ __forceinline__ float bperm(int idxB, float v) {
  return __int_as_float(__builtin_amdgcn_ds_bpermute(idxB, __float_as_int(v)));
}

// result lane l (<16): X[idx/4]; lane l (>=16): Y[idx/4]
__device__ __forceinline__ float cmb(float X, float Y, int idx, bool lo) {
  float rx = bperm(idx, X);
  float ry = bperm(idx, Y);
  return lo ? rx : ry;
}

__device__ __forceinline__ v2f mk2(float a, float b) { v2f v; v.x = a; v.y = b; return v; }

__device__ __forceinline__ v8f zero8() {
  v8f z;
#pragma unroll
  for (int i = 0; i < 8; ++i) z[i] = 0.0f;
  return z;
}

// D = A(16x4) * B(4x16) + C(16x16), all f32. 8-arg form:
// (neg_a, A, neg_b, B, c_mod, C, reuse_a, reuse_b)
__device__ __forceinline__ v8f wmma4(v2f a, v2f b, v8f c) {
  return __builtin_amdgcn_wmma_f32_16x16x4_f32(false, a, false, b, (short)0, c,
                                               false, false);
}

// LIF update (elementwise on D layout). reset uses the *incoming* mem.
__device__ __forceinline__ void lif(v8f& mem, const v8f cur, v8f& spk) {
#pragma unroll
  for (int i = 0; i < 8; ++i) {
    float r = (mem[i] > 1.0f) ? 1.0f : 0.0f;         // spike(mem - THR), THR=1
    float m = 0.9f * mem[i] + cur[i] - r;            // BETA=0.9, reset*THR
    mem[i] = m;
    spk[i] = (m > 1.0f) ? 1.0f : 0.0f;               // spike(mem_new - THR)
  }
}

// Augmented weight element: cols [0,ind) = W, col ind = bias, else 0; rows >= outd -> 0
__device__ __forceinline__ float welem(const float* W, const float* b,
                                       int outd, int ind, int row, int col) {
  if (row >= outd) return 0.0f;
  if (col < ind)   return W[row * ind + col];
  if (col == ind)  return b[row];
  return 0.0f;
}

// Build 4 B-operand chunks (rows 0..15) from a 16-neuron spike set in D layout.
__device__ __forceinline__ void bchunks(const v8f s, int IDXLO, int IDXHI,
                                        bool lo, v2f B[4]) {
  B[0] = mk2(cmb(s[0], s[2], IDXLO, lo), cmb(s[1], s[3], IDXLO, lo));
  B[1] = mk2(cmb(s[4], s[6], IDXLO, lo), cmb(s[5], s[7], IDXLO, lo));
  B[2] = mk2(cmb(s[0], s[2], IDXHI, lo), cmb(s[1], s[3], IDXHI, lo));
  B[3] = mk2(cmb(s[4], s[6], IDXHI, lo), cmb(s[5], s[7], IDXHI, lo));
}

__global__ __launch_bounds__(256, 1) void MultitaskSNN_kernel(
    const float* __restrict__ x,
    const float* __restrict__ WeC, const float* __restrict__ beC,
    const float* __restrict__ WeR, const float* __restrict__ beR,
    const float* __restrict__ Ws,  const float* __restrict__ bs,
    const float* __restrict__ Wch, const float* __restrict__ bch,
    const float* __restrict__ Wco, const float* __restrict__ bco,
    const float* __restrict__ Wrh, const float* __restrict__ brh,
    const float* __restrict__ Wro, const float* __restrict__ bro,
    float* __restrict__ out) {
  const int tid    = blockIdx.x * blockDim.x + threadIdx.x;
  const int lane   = threadIdx.x & 31;
  const int wave   = tid >> 5;
  const int b0     = wave * 16;            // 16 batch elements per wave
  const int lane16 = lane & 15;
  const bool lo    = lane < 16;
  const int hl     = lane >> 4;

  const int IDXLO = lane16 * 4;            // byte index for ds_bpermute, lanes 0..15
  const int IDXHI = IDXLO + 64;            // lanes 16..31 of source

  const int row = lane16;
  const int c0  = lo ? 0 : 2;              // A-layout K offsets within a chunk
  const int c1  = lo ? 1 : 3;

  // ---- Preload augmented weights in A layout (once per wave) ----
  // Encoder: rows 0-7 = We_c (class), rows 8-15 = We_r (reg); in=6, bias col 6.
  v2f A_enc[2];
#pragma unroll
  for (int c = 0; c < 2; ++c) {
    float ax, ay;
    if (row < 8) {
      ax = welem(WeC, beC, 8, 6, row, 4 * c + c0);
      ay = welem(WeC, beC, 8, 6, row, 4 * c + c1);
    } else {
      ax = welem(WeR, beR, 8, 6, row - 8, 4 * c + c0);
      ay = welem(WeR, beR, 8, 6, row - 8, 4 * c + c1);
    }
    A_enc[c] = mk2(ax, ay);
  }
  v2f A_sh0[5], A_sh1[5];                  // Ws 24x16 (+bias col 16): M tiles 0-15, 16-23
#pragma unroll
  for (int c = 0; c < 5; ++c) {
    A_sh0[c] = mk2(welem(Ws, bs, 24, 16, row,      4 * c + c0),
                   welem(Ws, bs, 24, 16, row,      4 * c + c1));
    A_sh1[c] = mk2(welem(Ws, bs, 24, 16, row + 16, 4 * c + c0),
                   welem(Ws, bs, 24, 16, row + 16, 4 * c + c1));
  }
  v2f A_ch[7], A_rh[7];                    // 16x24 (+bias col 24)
#pragma unroll
  for (int c = 0; c < 7; ++c) {
    A_ch[c] = mk2(welem(Wch, bch, 16, 24, row, 4 * c + c0),
                  welem(Wch, bch, 16, 24, row, 4 * c + c1));
    A_rh[c] = mk2(welem(Wrh, brh, 16, 24, row, 4 * c + c0),
                  welem(Wrh, brh, 16, 24, row, 4 * c + c1));
  }
  v2f A_co[5], A_ro[5];                    // 3x16, 1x16 (+bias col 16)
#pragma unroll
  for (int c = 0; c < 5; ++c) {
    A_co[c] = mk2(welem(Wco, bco, 3, 16, row, 4 * c + c0),
                  welem(Wco, bco, 3, 16, row, 4 * c + c1));
    A_ro[c] = mk2(welem(Wro, bro, 1, 16, row, 4 * c + c0),
                  welem(Wro, bro, 1, 16, row, 4 * c + c1));
  }

  const v2f BONE = mk2(lo ? 1.0f : 0.0f, 0.0f);  // bias row (=1) chunk of B

  // ---- Membrane states (D layout), zero-initialized ----
  v8f m_enc = zero8(), m_sh0 = zero8(), m_sh1 = zero8();
  v8f m_ch = zero8(), m_co = zero8(), m_rh = zero8(), m_ro = zero8();

  // ---- Output offsets: (m_co, s_ch, m_ro, s_rh, s_sh, s_ec, s_er) concat flat ----
  const size_t TB    = (size_t)TSTEPS * BATCH;
  const size_t bl    = (size_t)(b0 + lane16);
  const size_t off_mco  = TB * 0  + bl * 3;
  const size_t off_sch  = TB * 3  + bl * 16 + (size_t)hl * 8;
  const size_t off_mro  = TB * 19 + bl;
  const size_t off_srh  = TB * 20 + bl * 16 + (size_t)hl * 8;
  const size_t off_ssh  = TB * 36 + bl * 24 + (size_t)hl * 8;
  const size_t off_ssh1 = TB * 36 + bl * 24 + 16;            // rows 16-23 (lo lanes)
  const size_t off_sec  = TB * 60 + bl * 8;
  const size_t off_ser  = TB * 68 + bl * 8;
  const size_t xoff     = bl * 6;

  for (int t = 0; t < TSTEPS; ++t) {
    // ---- Encoder currents: [We_c;We_r] (16x7aug) @ [x;1] ----
    const float* xb = x + (size_t)t * (BATCH * 6) + xoff;
    v2f Bx0 = mk2(xb[c0], xb[c1]);                           // rows 0-3 = x0..x3
    float x4 = xb[4], x5 = xb[5];
    v2f Bx1 = mk2(lo ? x4 : 1.0f, lo ? x5 : 0.0f);           // rows 4,5,6(=1),7(=0)

    v8f cur = zero8();
    cur = wmma4(A_enc[0], Bx0, cur);
    cur = wmma4(A_enc[1], Bx1, cur);
    v8f s_enc;
    lif(m_enc, cur, s_enc);                                   // rows 0-7=s_ec, 8-15=s_er

    // ---- Shared layer: Ws @ [comb;1], comb = s_enc (16 neurons) ----
    v2f Be[4];
    bchunks(s_enc, IDXLO, IDXHI, lo, Be);
    v8f cur0 = zero8(), cur1 = zero8();
#pragma unroll
    for (int c = 0; c < 4; ++c) {
      cur0 = wmma4(A_sh0[c], Be[c], cur0);
      cur1 = wmma4(A_sh1[c], Be[c], cur1);
    }
    cur0 = wmma4(A_sh0[4], BONE, cur0);
    cur1 = wmma4(A_sh1[4], BONE, cur1);
    v8f s_sh0, s_sh1;
    lif(m_sh0, cur0, s_sh0);
    lif(m_sh1, cur1, s_sh1);                                  // neurons 16-23 in lo rows

    // ---- ch & rh share the s_sh B operand (24 neurons + bias row) ----
    v2f Bs[6];
    bchunks(s_sh0, IDXLO, IDXHI, lo, Bs);
    Bs[4] = mk2(cmb(s_sh1[0], s_sh1[2], IDXLO, lo), cmb(s_sh1[1], s_sh1[3], IDXLO, lo));
    Bs[5] = mk2(cmb(s_sh1[4], s_sh1[6], IDXLO, lo), cmb(s_sh1[5], s_sh1[7], IDXLO, lo));
    v8f curch = zero8(), currh = zero8();
#pragma unroll
    for (int c = 0; c < 6; ++c) {
      curch = wmma4(A_ch[c], Bs[c], curch);
      currh = wmma4(A_rh[c], Bs[c], currh);
    }
    curch = wmma4(A_ch[6], BONE, curch);
    currh = wmma4(A_rh[6], BONE, currh);
    v8f s_ch, s_rh;
    lif(m_ch, curch, s_ch);
    lif(m_rh, currh, s_rh);

    // ---- Class output: Wco @ [s_ch;1] (only m_co needed) ----
    v2f Bc[4];
    bchunks(s_ch, IDXLO, IDXHI, lo, Bc);
    v8f curco = zero8();
#pragma unroll
    for (int c = 0; c < 4; ++c) curco = wmma4(A_co[c], Bc[c], curco);
    curco = wmma4(A_co[4], BONE, curco);
    v8f s_unused0;
    lif(m_co, curco, s_unused0);

    // ---- Regression output: Wro @ [s_rh;1] (only m_ro needed) ----
    v2f Br[4];
    bchunks(s_rh, IDXLO, IDXHI, lo, Br);
    v8f curro = zero8();
#pragma unroll
    for (int c = 0; c < 4; ++c) curro = wmma4(A_ro[c], Br[c], curro);
    curro = wmma4(A_ro[4], BONE, curro);
    v8f s_unused1;
    lif(m_ro, curro, s_unused1);

    // ---- Stores (D layout straight to global; rows fill cachelines) ----
    {
      float* p_sch = out + off_sch + (size_t)t * (BATCH * 16);
      float* p_srh = out + off_srh + (size_t)t * (BATCH * 16);
      float* p_ssh = out + off_ssh + (size_t)t * (BATCH * 24);
#pragma unroll
      for (int r = 0; r < 8; ++r) {
        p_sch[r] = s_ch[r];
        p_srh[r] = s_rh[r];
        p_ssh[r] = s_sh0[r];
      }
      if (lo) {
        float* p1  = out + off_ssh1 + (size_t)t * (BATCH * 24);
        float* pm  = out + off_mco + (size_t)t * (BATCH * 3);
        float* pr  = out + off_mro + (size_t)t * BATCH;
        float* pe  = out + off_sec + (size_t)t * (BATCH * 8);
#pragma unroll
        for (int r = 0; r < 8; ++r) p1[r] = s_sh1[r];
        pm[0] = m_co[0]; pm[1] = m_co[1]; pm[2] = m_co[2];
        pr[0] = m_ro[0];
#pragma unroll
        for (int r = 0; r < 8; ++r) pe[r] = s_enc[r];
      } else {
        float* pe = out + off_ser + (size_t)t * (BATCH * 8);
#pragma unroll
        for (int r = 0; r < 8; ++r) pe[r] = s_enc[r];       // hi lanes: rows 8-15 = s_er
      }
    }
  }
}

extern "C" void kernel_launch(void* const* d_in, const int* in_sizes, int n_in,
                              void* d_out, int out_size, void* d_ws, size_t ws_size,
                              hipStream_t stream) {
  (void)in_sizes; (void)n_in; (void)out_size; (void)d_ws; (void)ws_size;
  // 16384 batch / 16 per wave = 1024 waves; 8 waves (256 thr) per block -> 128 blocks
  MultitaskSNN_kernel<<<dim3(BATCH / 128), dim3(256), 0, stream>>>(
      (const float*)d_in[0],
      (const float*)d_in[1],  (const float*)d_in[2],
      (const float*)d_in[3],  (const float*)d_in[4],
      (const float*)d_in[5],  (const float*)d_in[6],
      (const float*)d_in[7],  (const float*)d_in[8],
      (const float*)d_in[9],  (const float*)d_in[10],
      (const float*)d_in[11], (const float*)d_in[12],
      (const float*)d_in[13], (const float*)d_in[14],
      (float*)d_out);
}